// MGN_loss_7438883356865
// MI455X (gfx1250) — compile-verified
//
#include <hip/hip_runtime.h>
#include <math.h>

// Problem constants (match reference)
#define HH    8
#define NN    2048
#define CC    4096
#define TT    3
#define DD    2048
#define MARGIN1 1.2f

// Gram kernel tiling
#define NB      64              // number of 32-wide row/col blocks (2048/32)
#define KC      64              // K-chunk staged in LDS per double-buffer step
#define NCHUNK  (DD / KC)       // 32
#define STR     68              // padded dwords per LDS row (64 data + 4 pad) -> bank-conflict-free
#define BUFDW   (32 * STR)      // 2176 dwords per buffer
#define BUFBYTES (BUFDW * 4)    // 8704 bytes

typedef __attribute__((ext_vector_type(2))) float v2f;
typedef __attribute__((ext_vector_type(8))) float v8f;

// ---------------------------------------------------------------------------
// Workspace init: ap = 0 (max accumulator, d >= 0), an = +inf bits, scalars = 0
// ---------------------------------------------------------------------------
__global__ void init_ws_kernel(unsigned* ap, unsigned* an, float* loss_acc,
                               unsigned* correct) {
    int i = blockIdx.x * blockDim.x + threadIdx.x;
    if (i < TT * NN) {
        ap[i] = 0u;           // float 0.0 bits
        an[i] = 0x7F800000u;  // +inf bits
    }
    if (i == 0) {
        *loss_acc = 0.0f;
        *correct  = 0u;
    }
}

// ---------------------------------------------------------------------------
// sq[t][n] = sum_k f[t][n][k]^2
// ---------------------------------------------------------------------------
__global__ __launch_bounds__(256) void sq_kernel(const float* __restrict__ feats,
                                                 float* __restrict__ sq) {
    int n = blockIdx.x;
    int t = blockIdx.y;
    const float* row = feats + ((size_t)t * NN + n) * DD;
    float s = 0.0f;
    for (int c = threadIdx.x; c < DD; c += 256) {
        float x = row[c];
        s += x * x;
    }
    __shared__ float sm[256];
    sm[threadIdx.x] = s;
    __syncthreads();
    for (int off = 128; off > 0; off >>= 1) {
        if (threadIdx.x < off) sm[threadIdx.x] += sm[threadIdx.x + off];
        __syncthreads();
    }
    if (threadIdx.x == 0) sq[t * NN + n] = sm[0];
}

// ---------------------------------------------------------------------------
// Fused CE (all heads) + prec@1 (head 0).  One block per (h, n) row.
// ---------------------------------------------------------------------------
__global__ __launch_bounds__(256) void ce_kernel(const float* __restrict__ logits,
                                                 const int* __restrict__ targets,
                                                 float* __restrict__ loss_acc,
                                                 unsigned* __restrict__ correct) {
    int row = blockIdx.x;          // 0 .. H*N-1
    int h = row / NN;
    int n = row - h * NN;
    const float* L = logits + (size_t)row * CC;

    float v[16];
    float mx = -__builtin_inff();
    int   mi = 0;
    #pragma unroll
    for (int i = 0; i < 16; ++i) {
        int c = threadIdx.x + i * 256;
        float x = L[c];
        v[i] = x;
        if (x > mx) { mx = x; mi = c; }   // ascending c -> first max kept
    }

    __shared__ float smx[256];
    __shared__ int   smi[256];
    __shared__ float ssum[256];
    smx[threadIdx.x] = mx;
    smi[threadIdx.x] = mi;
    __syncthreads();
    for (int off = 128; off > 0; off >>= 1) {
        if (threadIdx.x < off) {
            float o  = smx[threadIdx.x + off];
            int   oi = smi[threadIdx.x + off];
            float c0 = smx[threadIdx.x];
            int   ci = smi[threadIdx.x];
            if (o > c0 || (o == c0 && oi < ci)) {
                smx[threadIdx.x] = o;
                smi[threadIdx.x] = oi;
            }
        }
        __syncthreads();
    }
    float rmax = smx[0];
    int   ridx = smi[0];

    float se = 0.0f;
    #pragma unroll
    for (int i = 0; i < 16; ++i) se += expf(v[i] - rmax);
    ssum[threadIdx.x] = se;
    __syncthreads();
    for (int off = 128; off > 0; off >>= 1) {
        if (threadIdx.x < off) ssum[threadIdx.x] += ssum[threadIdx.x + off];
        __syncthreads();
    }

    if (threadIdx.x == 0) {
        int tgt = targets[n];
        float xt = L[tgt];
        float nll = logf(ssum[0]) + rmax - xt;
        atomicAdd(loss_acc, nll * (1.0f / (float)NN));
        if (h == 0 && ridx == tgt) atomicAdd(correct, 1u);
    }
}

// ---------------------------------------------------------------------------
// Symmetric Gram-GEMM with LDS double-buffered async staging and fused
// dual-sided batch-hard epilogue.
//
// One wave owns a 32x32 tile (rb, cb), cb >= rb only.  K staged through LDS
// in KC=64 chunks via GLOBAL_LOAD_ASYNC_TO_LDS_B128 (coalesced: each
// instruction reads two contiguous 256B row chunks).  LDS rows padded to 68
// dwords so each ds_load_b64 fragment fetch is 64-bank conflict-free.
//
// Fragment layouts per ISA 7.12.2 (32-bit, wave32):
//   A 16x4 : lane l -> row (l&15), K = 2*(l>>4) + {0,1}
//   B 4x16 : lane l -> col (l&15), K = 2*(l>>4) + {0,1}
//   C 16x16: lane l -> col (l&15), VGPR v -> row v + 8*(l>>4)
// ---------------------------------------------------------------------------
__global__ __launch_bounds__(32) void gram_kernel(const float* __restrict__ feats,
                                                  const float* __restrict__ sq,
                                                  unsigned* __restrict__ ap,
                                                  unsigned* __restrict__ an) {
    const int rb = blockIdx.x;     // row block (32 rows)
    const int cb = blockIdx.y;     // col block (32 cols)
    const int t  = blockIdx.z;     // branch
    if (cb < rb) return;           // symmetry: upper triangle only (uniform exit)

    const int lane = threadIdx.x;
    const int half = lane >> 4;
    const int l15  = lane & 15;

    __shared__ float lds[4 * BUFDW];   // A buf0, A buf1, B buf0, B buf1
    const unsigned ldsbase = (unsigned)(uintptr_t)&lds[0];

    const float* f   = feats + (size_t)t * NN * DD;
    const float* sqt = sq + t * NN;
    const int m0 = rb * 32;
    const int n0 = cb * 32;

    // Per-lane global bases for async fills: instr i covers rows {2i, 2i+1},
    // lane half selects the row within the pair, l15 selects the 16B column chunk.
    const float* gA = f + (size_t)(m0 + half) * DD + l15 * 4;
    const float* gB = f + (size_t)(n0 + half) * DD + l15 * 4;
    const unsigned ldsRowByte = (unsigned)((half * STR + l15 * 4) * 4);

    float sqmA[8], sqmB[8];
    #pragma unroll
    for (int v = 0; v < 8; ++v) {
        sqmA[v] = sqt[m0 + v + 8 * half];
        sqmB[v] = sqt[m0 + 16 + v + 8 * half];
    }
    const float sqn0 = sqt[n0 + l15];
    const float sqn1 = sqt[n0 + 16 + l15];

    float rMaxA[8], rMinA[8], rMaxB[8], rMinB[8];
    #pragma unroll
    for (int v = 0; v < 8; ++v) {
        rMaxA[v] = 0.0f; rMinA[v] = __builtin_inff();
        rMaxB[v] = 0.0f; rMinB[v] = __builtin_inff();
    }
    float cMax0 = 0.0f, cMin0 = __builtin_inff();
    float cMax1 = 0.0f, cMin1 = __builtin_inff();

    v8f c00 = {}, c01 = {}, c10 = {}, c11 = {};

    // Async fill of one double-buffer slot (32 x GLOBAL_LOAD_ASYNC_TO_LDS_B128).
    auto fill = [&](int sel, int k0) {
        unsigned la = ldsbase + (unsigned)sel * BUFBYTES + ldsRowByte;
        unsigned lb = ldsbase + 2u * BUFBYTES + (unsigned)sel * BUFBYTES + ldsRowByte;
        const float* pa = gA + k0;
        const float* pb = gB + k0;
        #pragma unroll
        for (int i = 0; i < 16; ++i) {
            asm volatile("global_load_async_to_lds_b128 %0, %1, off"
                         :: "v"(la + (unsigned)(i * 2 * STR * 4)),
                            "v"((unsigned long long)(uintptr_t)(pa + (size_t)i * 2 * DD))
                         : "memory");
            asm volatile("global_load_async_to_lds_b128 %0, %1, off"
                         :: "v"(lb + (unsigned)(i * 2 * STR * 4)),
                            "v"((unsigned long long)(uintptr_t)(pb + (size_t)i * 2 * DD))
                         : "memory");
        }
    };

    fill(0, 0);
    for (int ch = 0; ch < NCHUNK; ++ch) {
        const int sel = ch & 1;
        if (ch + 1 < NCHUNK) {
            fill(sel ^ 1, (ch + 1) * KC);
            // 32 async ops just issued for the next buffer; wait until only
            // those remain -> current buffer is fully in LDS.
            asm volatile("s_wait_asynccnt 0x20" ::: "memory");
        } else {
            asm volatile("s_wait_asynccnt 0x0" ::: "memory");
        }

        const float* A  = lds + sel * BUFDW;
        const float* Bf = lds + 2 * BUFDW + sel * BUFDW;
        #pragma unroll 4
        for (int kk = 0; kk < KC / 4; ++kk) {
            const int o = kk * 4 + half * 2;
            v2f a0 = *(const v2f*)(A  + l15 * STR + o);
            v2f a1 = *(const v2f*)(A  + (16 + l15) * STR + o);
            v2f b0 = *(const v2f*)(Bf + l15 * STR + o);
            v2f b1 = *(const v2f*)(Bf + (16 + l15) * STR + o);
            c00 = __builtin_amdgcn_wmma_f32_16x16x4_f32(false, a0, false, b0, (short)0, c00, false, false);
            c01 = __builtin_amdgcn_wmma_f32_16x16x4_f32(false, a0, false, b1, (short)0, c01, false, false);
            c10 = __builtin_amdgcn_wmma_f32_16x16x4_f32(false, a1, false, b0, (short)0, c10, false, false);
            c11 = __builtin_amdgcn_wmma_f32_16x16x4_f32(false, a1, false, b1, (short)0, c11, false, false);
        }
    }

    // Dual-sided epilogue: each element folds into its row's running min/max
    // (cross-lane) and its column's running min/max (in-lane).
    auto epi = [&](const v8f& c, const float* sqm, int mb, float sqn, int colg,
                   float* rMax, float* rMin, float& cMax, float& cMin) {
        #pragma unroll
        for (int v = 0; v < 8; ++v) {
            int m = mb + v + 8 * half;
            float d2 = sqm[v] + sqn - 2.0f * c[v];
            float d  = sqrtf(fmaxf(d2, 1e-12f));
            bool pos = ((m >> 2) == colg);
            float dp = pos ? d : 0.0f;
            float dn = pos ? __builtin_inff() : d;
            rMax[v] = fmaxf(rMax[v], dp);
            rMin[v] = fminf(rMin[v], dn);
            cMax = fmaxf(cMax, dp);
            cMin = fminf(cMin, dn);
        }
    };
    epi(c00, sqmA, m0,      sqn0, (n0 + l15) >> 2,      rMaxA, rMinA, cMax0, cMin0);
    epi(c01, sqmA, m0,      sqn1, (n0 + 16 + l15) >> 2, rMaxA, rMinA, cMax1, cMin1);
    epi(c10, sqmB, m0 + 16, sqn0, (n0 + l15) >> 2,      rMaxB, rMinB, cMax0, cMin0);
    epi(c11, sqmB, m0 + 16, sqn1, (n0 + 16 + l15) >> 2, rMaxB, rMinB, cMax1, cMin1);

    unsigned* apt = ap + t * NN;
    unsigned* ant = an + t * NN;

    // Row side: reduce across the 16 lanes holding one row, 1 atomic pair/row.
    #pragma unroll
    for (int v = 0; v < 8; ++v) {
        float mxA = rMaxA[v], mnA = rMinA[v];
        float mxB = rMaxB[v], mnB = rMinB[v];
        for (int off = 8; off > 0; off >>= 1) {
            mxA = fmaxf(mxA, __shfl_xor(mxA, off, 16));
            mnA = fminf(mnA, __shfl_xor(mnA, off, 16));
            mxB = fmaxf(mxB, __shfl_xor(mxB, off, 16));
            mnB = fminf(mnB, __shfl_xor(mnB, off, 16));
        }
        if (l15 == 0) {
            int mA = m0 + v + 8 * half;
            int mB = m0 + 16 + v + 8 * half;
            atomicMax(&apt[mA], __float_as_uint(mxA));
            atomicMin(&ant[mA], __float_as_uint(mnA));
            atomicMax(&apt[mB], __float_as_uint(mxB));
            atomicMin(&ant[mB], __float_as_uint(mnB));
        }
    }

    // Column side (transpose contribution), skipped on diagonal blocks.
    if (cb != rb) {
        cMax0 = fmaxf(cMax0, __shfl_xor(cMax0, 16));
        cMin0 = fminf(cMin0, __shfl_xor(cMin0, 16));
        cMax1 = fmaxf(cMax1, __shfl_xor(cMax1, 16));
        cMin1 = fminf(cMin1, __shfl_xor(cMin1, 16));
        if (lane < 16) {
            atomicMax(&apt[n0 + l15],      __float_as_uint(cMax0));
            atomicMin(&ant[n0 + l15],      __float_as_uint(cMin0));
            atomicMax(&apt[n0 + 16 + l15], __float_as_uint(cMax1));
            atomicMin(&ant[n0 + 16 + l15], __float_as_uint(cMin1));
        }
    }
}

// ---------------------------------------------------------------------------
// Final combine: triplet margin-ranking mean + CE sum -> out[0]; prec -> out[1]
// ---------------------------------------------------------------------------
__global__ __launch_bounds__(256) void final_kernel(const unsigned* __restrict__ ap,
                                                    const unsigned* __restrict__ an,
                                                    const float* __restrict__ loss_acc,
                                                    const unsigned* __restrict__ correct,
                                                    float* __restrict__ out) {
    float s = 0.0f;
    for (int i = threadIdx.x; i < TT * NN; i += 256) {
        float dap = __uint_as_float(ap[i]);
        float dan = __uint_as_float(an[i]);
        s += fmaxf(MARGIN1 + dap - dan, 0.0f);
    }
    __shared__ float sm[256];
    sm[threadIdx.x] = s;
    __syncthreads();
    for (int off = 128; off > 0; off >>= 1) {
        if (threadIdx.x < off) sm[threadIdx.x] += sm[threadIdx.x + off];
        __syncthreads();
    }
    if (threadIdx.x == 0) {
        out[0] = *loss_acc + sm[0] * (1.0f / (float)NN);
        out[1] = (float)(*correct) * (100.0f / (float)NN);
    }
}

// ---------------------------------------------------------------------------
extern "C" void kernel_launch(void* const* d_in, const int* in_sizes, int n_in,
                              void* d_out, int out_size, void* d_ws, size_t ws_size,
                              hipStream_t stream) {
    const float* logits  = (const float*)d_in[0];   // [8, 2048, 4096]
    const float* feats   = (const float*)d_in[1];   // [3, 2048, 2048]
    const int*   targets = (const int*)d_in[2];     // [2048]
    float* out = (float*)d_out;                     // [loss, prec]

    float*    ws       = (float*)d_ws;
    float*    loss_acc = ws;                         // 1
    unsigned* correct  = (unsigned*)(ws + 1);        // 1
    unsigned* ap       = (unsigned*)(ws + 2);        // T*N
    unsigned* an       = ap + TT * NN;               // T*N
    float*    sqv      = (float*)(an + TT * NN);     // T*N

    init_ws_kernel<<<(TT * NN + 255) / 256, 256, 0, stream>>>(ap, an, loss_acc, correct);
    sq_kernel<<<dim3(NN, TT), 256, 0, stream>>>(feats, sqv);
    ce_kernel<<<HH * NN, 256, 0, stream>>>(logits, targets, loss_acc, correct);
    gram_kernel<<<dim3(NB, NB, TT), 32, 0, stream>>>(feats, sqv, ap, an);
    final_kernel<<<1, 256, 0, stream>>>(ap, an, loss_acc, correct, out);
}